// GATHeadLayer_5351529251499
// MI455X (gfx1250) — compile-verified
//
#include <hip/hip_runtime.h>
#include <math.h>

#define N_NODES 100000
#define N_EDGES 1600000
#define IN_DIM  256
#define OUT_DIM 64
#define LEAKY   0.01f

typedef __attribute__((ext_vector_type(2))) float v2f;
typedef __attribute__((ext_vector_type(8))) float v8f;

// ---------- helpers ----------

__device__ __forceinline__ void atomic_fadd_relaxed(float* p, float v) {
    // lowers to global_atomic_add_f32 (no-return form since result unused)
    __hip_atomic_fetch_add(p, v, __ATOMIC_RELAXED, __HIP_MEMORY_SCOPE_AGENT);
}

__device__ __forceinline__ void atomic_fmax(float* p, float v) {
    // classic IEEE-754 ordered-bits trick; location initialized to -inf
    if (v >= 0.0f) atomicMax((int*)p, __float_as_int(v));
    else           atomicMin((unsigned int*)p, __float_as_uint(v));
}

// ---------- kernels ----------

// out[i]=0 for i<N*64 ; m=-inf, s=0 for i<N
__global__ void k_init(float* __restrict__ out, float* __restrict__ mx,
                       float* __restrict__ ssum) {
    int i = blockIdx.x * blockDim.x + threadIdx.x;
    if (i < N_NODES * OUT_DIM) out[i] = 0.0f;
    if (i < N_NODES) { mx[i] = -__builtin_inff(); ssum[i] = 0.0f; }
}

// z = h @ W   (fp32 WMMA 16x16x4; one wave -> 16x64 tile of z)
__global__ void __launch_bounds__(256)
k_gemm(const float* __restrict__ h, const float* __restrict__ W,
       float* __restrict__ z) {
    __shared__ float Wl[IN_DIM * OUT_DIM]; // 64 KB, fits WGP LDS
    for (int i = threadIdx.x; i < IN_DIM * OUT_DIM; i += 256) Wl[i] = W[i];
    __syncthreads();

    const int wave = blockIdx.x * 8 + (threadIdx.x >> 5);
    if (wave >= N_NODES / 16) return;

    const int lane = threadIdx.x & 31;
    const int m    = lane & 15;   // A row / B col within tile
    const int kh   = lane >> 4;   // K half-select (0 or 1)
    const int row0 = wave * 16;

    const float* arow = h + (size_t)(row0 + m) * IN_DIM + 2 * kh;

    v8f acc0 = {}, acc1 = {}, acc2 = {}, acc3 = {};

    #pragma unroll 4
    for (int kk = 0; kk < IN_DIM; kk += 4) {
        // A[16x4] : lane holds A[m][kk + 2*kh + j], j=0,1  (aligned float2)
        v2f av = *(const v2f*)(arow + kk);
        // B[4x16] : lane holds B[kk + 2*kh + j][ntile*16 + m]
        const float* wp = Wl + (size_t)(kk + 2 * kh) * OUT_DIM + m;
        v2f b0 = { wp[ 0], wp[ 0 + OUT_DIM] };
        v2f b1 = { wp[16], wp[16 + OUT_DIM] };
        v2f b2 = { wp[32], wp[32 + OUT_DIM] };
        v2f b3 = { wp[48], wp[48 + OUT_DIM] };
        acc0 = __builtin_amdgcn_wmma_f32_16x16x4_f32(false, av, false, b0, (short)0, acc0, false, false);
        acc1 = __builtin_amdgcn_wmma_f32_16x16x4_f32(false, av, false, b1, (short)0, acc1, false, false);
        acc2 = __builtin_amdgcn_wmma_f32_16x16x4_f32(false, av, false, b2, (short)0, acc2, false, false);
        acc3 = __builtin_amdgcn_wmma_f32_16x16x4_f32(false, av, false, b3, (short)0, acc3, false, false);
    }

    // D layout: vgpr r, lanes 0-15 -> M=r, lanes 16-31 -> M=r+8; N = lane&15
    float* zrow = z + (size_t)(row0 + 8 * kh) * OUT_DIM + m;
    #pragma unroll
    for (int r = 0; r < 8; ++r) {
        zrow[(size_t)r * OUT_DIM +  0] = acc0[r];
        zrow[(size_t)r * OUT_DIM + 16] = acc1[r];
        zrow[(size_t)r * OUT_DIM + 32] = acc2[r];
        zrow[(size_t)r * OUT_DIM + 48] = acc3[r];
    }
}

// el[i] = z[i,:]·a[0:64],  er[i] = z[i,:]·a[64:128]   (warp per node)
__global__ void k_elr(const float* __restrict__ z, const float* __restrict__ a,
                      float* __restrict__ el, float* __restrict__ er) {
    int node = blockIdx.x * (blockDim.x >> 5) + (threadIdx.x >> 5);
    int lane = threadIdx.x & 31;
    if (node >= N_NODES) return;
    float z0 = z[(size_t)node * OUT_DIM + lane];
    float z1 = z[(size_t)node * OUT_DIM + 32 + lane];
    float pl = z0 * a[lane]      + z1 * a[32 + lane];
    float pr = z0 * a[64 + lane] + z1 * a[96 + lane];
    #pragma unroll
    for (int off = 16; off > 0; off >>= 1) {
        pl += __shfl_xor(pl, off, 32);
        pr += __shfl_xor(pr, off, 32);
    }
    if (lane == 0) { el[node] = pl; er[node] = pr; }
}

// pass 1: e = leaky_relu(el[src]+er[dst]); segment max into mx[dst]
__global__ void k_edge1(const float* __restrict__ el, const float* __restrict__ er,
                        const int* __restrict__ src, const int* __restrict__ dst,
                        float* __restrict__ ev, float* __restrict__ mx) {
    int e = blockIdx.x * blockDim.x + threadIdx.x;
    if (e >= N_EDGES) return;
    float v = el[src[e]] + er[dst[e]];
    v = (v > 0.0f) ? v : LEAKY * v;
    ev[e] = v;
    atomic_fmax(&mx[dst[e]], v);
}

// pass 2: ex = exp(e - m[dst]); segment sum into ssum[dst]
__global__ void k_edge2(const int* __restrict__ dst, const float* __restrict__ mx,
                        float* __restrict__ ev, float* __restrict__ ssum) {
    int e = blockIdx.x * blockDim.x + threadIdx.x;
    if (e >= N_EDGES) return;
    int d = dst[e];
    float ex = __expf(ev[e] - mx[d]);
    ev[e] = ex;
    atomic_fadd_relaxed(&ssum[d], ex);
}

// pass 3: out[dst,:] += (ex/ssum[dst]) * z[src,:]   (wave per edge)
__global__ void k_edge3(const int* __restrict__ src, const int* __restrict__ dst,
                        const float* __restrict__ ev, const float* __restrict__ ssum,
                        const float* __restrict__ z, float* __restrict__ out) {
    int e = blockIdx.x * (blockDim.x >> 5) + (threadIdx.x >> 5);
    if (e >= N_EDGES) return;
    int lane = threadIdx.x & 31;
    int s = src[e], d = dst[e];
    float alpha = ev[e] / ssum[d];
    float z0 = z[(size_t)s * OUT_DIM + lane];
    float z1 = z[(size_t)s * OUT_DIM + 32 + lane];
    atomic_fadd_relaxed(&out[(size_t)d * OUT_DIM + lane],      alpha * z0);
    atomic_fadd_relaxed(&out[(size_t)d * OUT_DIM + 32 + lane], alpha * z1);
}

// final: in-place ELU
__global__ void k_elu(float* __restrict__ out) {
    int i = blockIdx.x * blockDim.x + threadIdx.x;
    if (i >= N_NODES * OUT_DIM) return;
    float x = out[i];
    out[i] = (x > 0.0f) ? x : expm1f(x);
}

// ---------- launch ----------

extern "C" void kernel_launch(void* const* d_in, const int* in_sizes, int n_in,
                              void* d_out, int out_size, void* d_ws, size_t ws_size,
                              hipStream_t stream) {
    const float* h    = (const float*)d_in[0];
    const float* W    = (const float*)d_in[1];
    const float* a    = (const float*)d_in[2];
    const int*   esrc = (const int*)d_in[3];
    const int*   edst = (const int*)d_in[4];
    float* out = (float*)d_out;

    float* ws   = (float*)d_ws;
    float* z    = ws;                              // N*64 floats (25.6 MB)
    float* el   = z  + (size_t)N_NODES * OUT_DIM;  // N
    float* er   = el + N_NODES;                    // N
    float* mx   = er + N_NODES;                    // N
    float* ssum = mx + N_NODES;                    // N
    float* ev   = ssum + N_NODES;                  // E (6.4 MB)

    const int nelem = N_NODES * OUT_DIM;

    k_init <<<(nelem + 255) / 256, 256, 0, stream>>>(out, mx, ssum);
    k_gemm <<<(N_NODES / 16 + 7) / 8, 256, 0, stream>>>(h, W, z);
    k_elr  <<<(N_NODES + 7) / 8, 256, 0, stream>>>(z, a, el, er);
    k_edge1<<<(N_EDGES + 255) / 256, 256, 0, stream>>>(el, er, esrc, edst, ev, mx);
    k_edge2<<<(N_EDGES + 255) / 256, 256, 0, stream>>>(edst, mx, ev, ssum);
    k_edge3<<<(N_EDGES + 7) / 8, 256, 0, stream>>>(esrc, edst, ev, ssum, z, out);
    k_elu  <<<(nelem + 255) / 256, 256, 0, stream>>>(out);
}